// Conv2D_33406255629046
// MI455X (gfx1250) — compile-verified
//
#include <hip/hip_runtime.h>

// CDNA5 / gfx1250: wave32, WMMA 16x16 shapes, 320KB LDS per WGP.
typedef __attribute__((ext_vector_type(2))) float v2f;
typedef __attribute__((ext_vector_type(8))) float v8f;

#define IN_N   4096
#define OUT_N  4090
#define KW     7

#define WAVES   8                 // 256 threads / wave32
#define BROWS   (WAVES * 16)      // 128 output rows per block
#define LROWS   (BROWS + KW - 1)  // 134 input rows staged
#define LCOLS   24                // 16 outputs + 6 halo, padded to 24 (K pad; B=0 there)
#define LSTRIDE 26                // even -> 8B-aligned ds_load_b64; conflict-free column reads
#define NKC     6                 // K = 24 in chunks of 4
#define NBFRAG  (KW * NKC * 32)   // precomputed B fragments: (dy, kc, lane)

// Conv as banded matmul per filter row dy:
//   O[m,n] += sum_c X[y0+dy+m, x0+c] * B_dy[c,n],  B_dy[c,n] = W[dy, c-n] for 0<=c-n<=6 else 0.
// WMMA f32 16x16x4 fragment coords (wave32):
//   A 16x4: lanes 0-15 -> M=lane, K={0,1} in regs {0,1}; lanes 16-31 -> K={2,3}.
//   C/D:    reg r, lane l -> row (l>>4)*8 + r, col l&15.
__global__ __launch_bounds__(256) void conv7x7_wmma_f32(
    const float* __restrict__ x, const float* __restrict__ w,
    const float* __restrict__ bias, float* __restrict__ out) {
  __shared__ __align__(16) float sX[LROWS * LSTRIDE];
  __shared__ __align__(16) v2f   sB[NBFRAG];   // B fragments, tile-invariant

  const int tid  = threadIdx.x;
  const int lane = tid & 31;
  const int wv   = tid >> 5;            // wave id 0..7 -> vertical 16-row slice

  const int col0 = blockIdx.x * 16;     // output tile column base
  const int row0 = blockIdx.y * BROWS;  // output tile row base (block)

  // ---- Stage input halo tile (clamped at image edge) ----
  for (int i = tid; i < LROWS * LCOLS; i += 256) {
    const int r = i / LCOLS, c = i % LCOLS;
    int gr = row0 + r; if (gr > IN_N - 1) gr = IN_N - 1;
    int gc = col0 + c; if (gc > IN_N - 1) gc = IN_N - 1;
    sX[r * LSTRIDE + c] = x[gr * IN_N + gc];
  }

  // ---- Build banded-weight B fragment table once per block ----
  for (int i = tid; i < NBFRAG; i += 256) {
    const int dy = i / (NKC * 32);
    const int kc = (i / 32) % NKC;
    const int ln = i & 31;
    const int n  = ln & 15;
    const int c0 = kc * 4 + (ln >> 4) * 2;
    const int d0 = c0 - n;               // filter tap index for reg0
    v2f bb;
    bb.x = (d0 >=  0 && d0 <= 6) ? w[dy * KW + d0]     : 0.0f;
    bb.y = (d0 >= -1 && d0 <= 5) ? w[dy * KW + d0 + 1] : 0.0f;
    sB[i] = bb;
  }
  __syncthreads();

  const int mn    = lane & 15;   // A row (M) == D column (N)
  const int khalf = lane >> 4;

  v8f acc = {};                  // fp32 accumulators for the 16x16 tile

#pragma unroll
  for (int dy = 0; dy < KW; ++dy) {
    const float* aptr = &sX[(wv * 16 + dy + mn) * LSTRIDE + khalf * 2];
#pragma unroll
    for (int kc = 0; kc < NKC; ++kc) {
      const v2f a = *reinterpret_cast<const v2f*>(aptr + kc * 4);  // ds_load_b64
      const v2f b = sB[(dy * NKC + kc) * 32 + lane];               // ds_load_b64
      acc = __builtin_amdgcn_wmma_f32_16x16x4_f32(
          /*neg_a=*/false, a, /*neg_b=*/false, b,
          /*c_mod=*/(short)0, acc, /*reuse_a=*/false, /*reuse_b=*/false);
    }
  }

  // ---- Epilogue: add bias, guarded store of the 16x16 tile ----
  const float bs = bias[0];
#pragma unroll
  for (int r = 0; r < 8; ++r) {
    const int orow = row0 + wv * 16 + khalf * 8 + r;
    const int ocol = col0 + mn;
    if (orow < OUT_N && ocol < OUT_N)
      out[orow * OUT_N + ocol] = acc[r] + bs;
  }
}

extern "C" void kernel_launch(void* const* d_in, const int* in_sizes, int n_in,
                              void* d_out, int out_size, void* d_ws, size_t ws_size,
                              hipStream_t stream) {
  const float* x    = (const float*)d_in[0];
  const float* w    = (const float*)d_in[1];
  const float* bias = (const float*)d_in[2];
  float* out        = (float*)d_out;

  dim3 grid((OUT_N + 15) / 16, (OUT_N + BROWS - 1) / BROWS);
  conv7x7_wmma_f32<<<grid, dim3(256), 0, stream>>>(x, w, bias, out);
}